// ViewTranformer_3496103378973
// MI455X (gfx1250) — compile-verified
//
#include <hip/hip_runtime.h>
#include <math.h>

#define B_  16
#define N_  8192
#define D_  512

typedef float v2f __attribute__((ext_vector_type(2)));
typedef float v8f __attribute__((ext_vector_type(8)));

// ---------------- block reduction helpers ----------------
template<int BS>
__device__ inline float blk_sum(float v, float* red) {
  int t = threadIdx.x;
  red[t] = v; __syncthreads();
  for (int s = BS/2; s > 0; s >>= 1) {
    if (t < s) red[t] += red[t+s];
    __syncthreads();
  }
  float r = red[0]; __syncthreads();
  return r;
}
template<int BS>
__device__ inline float blk_max(float v, float* red) {
  int t = threadIdx.x;
  red[t] = v; __syncthreads();
  for (int s = BS/2; s > 0; s >>= 1) {
    if (t < s) red[t] = fmaxf(red[t], red[t+s]);
    __syncthreads();
  }
  float r = red[0]; __syncthreads();
  return r;
}

// ---------------- K1: masked centroid + voting ratio ----------------
__global__ void __launch_bounds__(256) k1_center(
    const float* __restrict__ xyz, const float* __restrict__ mask,
    float* __restrict__ center, float* __restrict__ vr) {
  __shared__ float red[256];
  int row = blockIdx.x;            // b*4 + j
  int b = row >> 2;
  const float* mk = mask + (size_t)row * N_;
  const float* xb = xyz + (size_t)b * 3 * N_;
  float cnt = 0.f, sx = 0.f, sy = 0.f, sz = 0.f;
  for (int n = threadIdx.x; n < N_; n += 256) {
    float m = mk[n];
    cnt += m;
    sx += m * xb[n];
    sy += m * xb[N_ + n];
    sz += m * xb[2*N_ + n];
  }
  cnt = blk_sum<256>(cnt, red);
  sx  = blk_sum<256>(sx,  red);
  sy  = blk_sum<256>(sy,  red);
  sz  = blk_sum<256>(sz,  red);
  if (threadIdx.x == 0) {
    float d = fmaxf(cnt, 1.0f);
    center[row*3+0] = sx / d;
    center[row*3+1] = sy / d;
    center[row*3+2] = sz / d;
    vr[row] = cnt * (1.0f / (float)N_);
  }
}

// ---------------- K2: MLP + q-projection + Wk fold ----------------
__global__ void __launch_bounds__(512) k2_mlp(
    const float* __restrict__ center,
    const float* __restrict__ W1, const float* __restrict__ b1,
    const float* __restrict__ g1, const float* __restrict__ be1,
    const float* __restrict__ W2, const float* __restrict__ b2,
    const float* __restrict__ g2, const float* __restrict__ be2,
    const float* __restrict__ Wq, const float* __restrict__ bq,
    const float* __restrict__ Wk, const float* __restrict__ bk,
    float* __restrict__ qk_g, float* __restrict__ qbias_g) {
  __shared__ float abuf[512];
  __shared__ float qbuf[512];
  __shared__ float red[512];
  int row = blockIdx.x; int b = row >> 2; int i = row & 3;
  int col = threadIdx.x;
  float c0 = center[row*3+0], c1 = center[row*3+1], c2 = center[row*3+2];

  // Linear(3,D) + LN + ReLU
  float t = c0*W1[col] + c1*W1[512+col] + c2*W1[1024+col] + b1[col];
  float mu  = blk_sum<512>(t, red) * (1.0f/512.0f);
  float dv  = t - mu;
  float var = blk_sum<512>(dv*dv, red) * (1.0f/512.0f);
  float h = dv * rsqrtf(var + 1e-5f) * g1[col] + be1[col];
  h = fmaxf(h, 0.0f);
  abuf[col] = h; __syncthreads();

  // Linear(D,D) + LN
  float t2 = b2[col];
  for (int d = 0; d < 512; ++d) t2 += abuf[d] * W2[(size_t)d*512 + col];
  mu  = blk_sum<512>(t2, red) * (1.0f/512.0f);
  dv  = t2 - mu;
  var = blk_sum<512>(dv*dv, red) * (1.0f/512.0f);
  float h3 = dv * rsqrtf(var + 1e-5f) * g2[col] + be2[col];
  abuf[col] = h3; __syncthreads();

  // q = h3 @ Wq + bq
  float q = bq[col];
  for (int d = 0; d < 512; ++d) q += abuf[d] * Wq[(size_t)d*512 + col];
  qbuf[col] = q; __syncthreads();

  // fold with Wk per head: qk[b, h*4+i, col] = sum_e q[h*128+e] * Wk[col, h*128+e]
  const float scale = 0.0883883476483184405f;  // 1/sqrt(128)
  for (int hh = 0; hh < 4; ++hh) {
    float acc = 0.f;
    const float* wkrow = Wk + (size_t)col*512 + hh*128;
    const float* qh = qbuf + hh*128;
    for (int e = 0; e < 128; ++e) acc += qh[e] * wkrow[e];
    qk_g[((size_t)(b*16 + hh*4 + i))*512 + col] = acc * scale;
  }
  if (col < 4) {
    float acc = 0.f;
    for (int e = 0; e < 128; ++e) acc += qbuf[col*128+e] * bk[col*128+e];
    qbias_g[b*16 + col*4 + i] = acc * scale;
  }
}

// ---------------- K3: voting attention bias, log-softmax over N ----------------
__global__ void __launch_bounds__(512) k3_aw(
    const float* __restrict__ mask, const float* __restrict__ vr,
    float* __restrict__ aw_g) {
  __shared__ float awb[N_];
  __shared__ float red[512];
  int b = blockIdx.x;
  const float* mb = mask + (size_t)b*4*N_;
  float v0 = vr[b*4+0], v1 = vr[b*4+1], v2 = vr[b*4+2], v3 = vr[b*4+3];
  float vcnt = 0.f;
  for (int n = threadIdx.x; n < N_; n += 512) {
    float m0 = mb[n], m1 = mb[N_+n], m2 = mb[2*N_+n], m3 = mb[3*N_+n];
    float s = m0+m1+m2+m3;
    float a = v0*m0 + v1*m1 + v2*m2 + v3*m3;
    bool valid = s > 0.f;
    awb[n] = valid ? a : -1e9f;
    vcnt += valid ? 1.f : 0.f;
  }
  float tot = blk_sum<512>(vcnt, red);
  if (tot <= 0.f) {
    for (int n = threadIdx.x; n < N_; n += 512) awb[n] = 0.f;
  }
  __syncthreads();
  float lm = -3.4e38f;
  for (int n = threadIdx.x; n < N_; n += 512) lm = fmaxf(lm, awb[n]);
  float mx = blk_max<512>(lm, red);
  float ls = 0.f;
  for (int n = threadIdx.x; n < N_; n += 512) ls += __expf(awb[n] - mx);
  float se = blk_sum<512>(ls, red);
  float lg = __logf(se) + mx;
  for (int n = threadIdx.x; n < N_; n += 512) aw_g[(size_t)b*N_ + n] = awb[n] - lg;
}

// ---------------- K4: WMMA score pass: s[b,hi,n] = qk . PF + qbias + aw ----------------
#define QKSTRIDE 516   // padded LDS stride (516 % 64 == 4 -> conflict-free A reads)
__global__ void __launch_bounds__(256) k4_scores(
    const float* __restrict__ PF, const float* __restrict__ qk_g,
    const float* __restrict__ qbias_g, const float* __restrict__ aw_g,
    float* __restrict__ scores) {
  __shared__ float qk_lds[16 * QKSTRIDE];
  int b = blockIdx.y;
  int wave = threadIdx.x >> 5;
  int lane = threadIdx.x & 31;
  int n0 = blockIdx.x * 128 + wave * 16;

  for (int idx = threadIdx.x; idx < 16*512; idx += 256) {
    int hi = idx >> 9; int d = idx & 511;
    qk_lds[hi*QKSTRIDE + d] = qk_g[((size_t)(b*16 + hi))*512 + d];
  }
  __syncthreads();

  int m    = lane & 15;          // A row (hi) / B column (n)
  int koff = (lane >> 4) * 2;    // lanes 0-15: K=0,1 ; lanes 16-31: K=2,3
  const float* PFb  = PF + (size_t)b * D_ * N_;
  const float* qrow = qk_lds + m * QKSTRIDE;

  v8f c = {};
  for (int k = 0; k < 512; k += 4) {
    v2f a, bb;
    a[0] = qrow[k + koff];
    a[1] = qrow[k + koff + 1];
    const float* p = PFb + (size_t)(k + koff) * N_ + n0 + m;
    bb[0] = p[0];
    bb[1] = p[N_];
    if (k < 512 - 16) __builtin_prefetch(p + (size_t)16 * N_, 0, 1);
    c = __builtin_amdgcn_wmma_f32_16x16x4_f32(false, a, false, bb,
                                              (short)0, c, false, false);
  }

  int nc = n0 + m;
  float awv = aw_g[(size_t)b*N_ + nc];
  int mbase = (lane >> 4) * 8;   // C layout: lanes>=16 hold rows M=8..15
  #pragma unroll
  for (int r = 0; r < 8; ++r) {
    int hi = mbase + r;
    scores[((size_t)(b*16 + hi))*N_ + nc] = c[r] + qbias_g[b*16 + hi] + awv;
  }
}

// ---------------- K5: row softmax over N (in place) ----------------
__global__ void __launch_bounds__(256) k5_softmax(float* __restrict__ scores) {
  __shared__ float ebuf[N_];
  __shared__ float red[256];
  int r = blockIdx.x;               // b*16 + hi
  float* base = scores + (size_t)r * N_;
  float lm = -3.4e38f;
  for (int n = threadIdx.x; n < N_; n += 256) lm = fmaxf(lm, base[n]);
  float mx = blk_max<256>(lm, red);
  float ls = 0.f;
  for (int n = threadIdx.x; n < N_; n += 256) {
    float e = __expf(base[n] - mx);
    ebuf[n] = e; ls += e;
  }
  float tot = blk_sum<256>(ls, red);
  float inv = 1.0f / tot;
  for (int n = threadIdx.x; n < N_; n += 256) base[n] = ebuf[n] * inv;
}

// ---------------- K6: u[b,hi,d] = sum_n attn[b,hi,n] * PF[b,d,n] ----------------
// attn chunks are staged into LDS with GLOBAL_LOAD_ASYNC_TO_LDS_B128 (ASYNCcnt),
// bypassing VGPRs entirely; FMA stream over PF stays in registers.
__global__ void __launch_bounds__(256) k6_u(
    const float* __restrict__ PF, const float* __restrict__ attn,
    float* __restrict__ u_g) {
  __shared__ float albuf[16 * 512];
  int b = blockIdx.y;
  int wave = threadIdx.x >> 5;
  int lane = threadIdx.x & 31;
  int d0 = blockIdx.x * 32 + wave * 4;
  const float* PFb = PF   + (size_t)b * D_ * N_;
  const float* atb = attn + (size_t)b * 16 * N_;
  float acc[4][16];
  #pragma unroll
  for (int dd = 0; dd < 4; ++dd)
    #pragma unroll
    for (int hi = 0; hi < 16; ++hi) acc[dd][hi] = 0.f;

  for (int nc = 0; nc < N_; nc += 512) {
    __syncthreads();   // all waves done reading previous chunk
    // async stage: 2048 x float4 = 16 rows x 512 cols
    for (int j = threadIdx.x; j < 2048; j += 256) {
      int hi = j >> 7;           // 128 float4 per row
      int v  = j & 127;
      unsigned lds_addr = (unsigned)(uintptr_t)&albuf[hi*512 + v*4];
      const float* gp = atb + (size_t)hi * N_ + nc + v*4;
      asm volatile("global_load_async_to_lds_b128 %0, %1, off"
                   :: "v"(lds_addr), "v"(gp) : "memory");
    }
    asm volatile("s_wait_asynccnt 0x0" ::: "memory");
    __syncthreads();   // all waves' async data visible
    for (int nn = lane; nn < 512; nn += 32) {
      float p0 = PFb[(size_t)(d0+0)*N_ + nc + nn];
      float p1 = PFb[(size_t)(d0+1)*N_ + nc + nn];
      float p2 = PFb[(size_t)(d0+2)*N_ + nc + nn];
      float p3 = PFb[(size_t)(d0+3)*N_ + nc + nn];
      #pragma unroll
      for (int hi = 0; hi < 16; ++hi) {
        float a = albuf[hi*512 + nn];
        acc[0][hi] += p0*a; acc[1][hi] += p1*a;
        acc[2][hi] += p2*a; acc[3][hi] += p3*a;
      }
    }
  }
  #pragma unroll
  for (int dd = 0; dd < 4; ++dd)
    #pragma unroll
    for (int hi = 0; hi < 16; ++hi) {
      float v = acc[dd][hi];
      for (int off = 16; off >= 1; off >>= 1) v += __shfl_xor(v, off, 32);
      acc[dd][hi] = v;
    }
  if (lane == 0) {
    #pragma unroll
    for (int dd = 0; dd < 4; ++dd)
      #pragma unroll
      for (int hi = 0; hi < 16; ++hi)
        u_g[((size_t)(b*16 + hi))*512 + d0 + dd] = acc[dd][hi];
  }
}

// ---------------- K7: out = concat_h(u_h @ Wv_h + bv_h) @ Wo + bo ----------------
__global__ void __launch_bounds__(256) k7_out(
    const float* __restrict__ u_g,
    const float* __restrict__ Wv, const float* __restrict__ bv,
    const float* __restrict__ Wo, const float* __restrict__ bo,
    float* __restrict__ out) {
  __shared__ float uu[4*512];
  __shared__ float ww[512];
  int row = blockIdx.x; int b = row >> 2; int i = row & 3;
  for (int idx = threadIdx.x; idx < 2048; idx += 256) {
    int hh = idx >> 9; int d = idx & 511;
    uu[idx] = u_g[((size_t)(b*16 + hh*4 + i))*512 + d];
  }
  __syncthreads();
  for (int e = threadIdx.x; e < 512; e += 256) {
    int hh = e >> 7;
    float acc = bv[e];
    const float* ub = uu + hh*512;
    for (int d = 0; d < 512; ++d) acc += ub[d] * Wv[(size_t)d*512 + e];
    ww[e] = acc;
  }
  __syncthreads();
  for (int f = threadIdx.x; f < 512; f += 256) {
    float acc = bo[f];
    for (int e = 0; e < 512; ++e) acc += ww[e] * Wo[(size_t)e*512 + f];
    out[(size_t)row*512 + f] = acc;
  }
}

// ---------------- launch ----------------
extern "C" void kernel_launch(void* const* d_in, const int* in_sizes, int n_in,
                              void* d_out, int out_size, void* d_ws, size_t ws_size,
                              hipStream_t stream) {
  (void)in_sizes; (void)n_in; (void)out_size; (void)ws_size;
  const float* xyz  = (const float*)d_in[0];
  const float* pf   = (const float*)d_in[1];
  const float* mask = (const float*)d_in[2];
  const float* W1   = (const float*)d_in[3];
  const float* b1   = (const float*)d_in[4];
  const float* g1   = (const float*)d_in[5];
  const float* be1  = (const float*)d_in[6];
  const float* W2   = (const float*)d_in[7];
  const float* b2   = (const float*)d_in[8];
  const float* g2   = (const float*)d_in[9];
  const float* be2  = (const float*)d_in[10];
  const float* Wq   = (const float*)d_in[11];
  const float* bq   = (const float*)d_in[12];
  const float* Wk   = (const float*)d_in[13];
  const float* bk   = (const float*)d_in[14];
  const float* Wv   = (const float*)d_in[15];
  const float* bv   = (const float*)d_in[16];
  const float* Wo   = (const float*)d_in[17];
  const float* bo   = (const float*)d_in[18];
  float* out = (float*)d_out;
  float* ws  = (float*)d_ws;

  // workspace layout (float offsets)
  float* center = ws;             // 192
  float* vr     = ws + 192;       // 64
  float* qbias  = ws + 256;       // 256
  float* qk     = ws + 512;       // 131072
  float* aw     = ws + 131584;    // 131072
  float* scores = ws + 262656;    // 2097152 (reused as attn)
  float* u      = ws + 2359808;   // 131072   (total ~9.96 MB)

  k1_center <<<64, 256, 0, stream>>>(xyz, mask, center, vr);
  k2_mlp    <<<64, 512, 0, stream>>>(center, W1,b1,g1,be1, W2,b2,g2,be2,
                                     Wq,bq, Wk,bk, qk, qbias);
  k3_aw     <<<B_, 512, 0, stream>>>(mask, vr, aw);
  k4_scores <<<dim3(N_/128, B_), 256, 0, stream>>>(pf, qk, qbias, aw, scores);
  k5_softmax<<<B_*16, 256, 0, stream>>>(scores);
  k6_u      <<<dim3(D_/32, B_), 256, 0, stream>>>(pf, scores, u);
  k7_out    <<<64, 256, 0, stream>>>(u, Wv, bv, Wo, bo, out);
}